// SGON1D_42906723287397
// MI455X (gfx1250) — compile-verified
//
#include <hip/hip_runtime.h>
#include <cstddef>

// Problem constants (must match reference)
#define BB 8
#define NN 2048
#define MM 256
#define QQ 16384
#define DD 32
#define RR 16
#define KK 16
#define HH 64
#define CG_ITERS 20
static constexpr float LAMBDA  = 5.0f;
static constexpr float RADIUSF = 1.5f / 256.0f;   // exactly representable (3/512)

typedef __attribute__((ext_vector_type(2))) float v2f;
typedef __attribute__((ext_vector_type(8))) float v8f;

// ---------------------------------------------------------------------------
// Kernel 1: global feature path
//   g = mean_n relu([xs,us] @ Wg1 + bg1);  gvec = g @ Wg2 + bg2   (per batch)
// ---------------------------------------------------------------------------
__global__ __launch_bounds__(256)
void sgon_gfeat_kernel(const float* __restrict__ xs, const float* __restrict__ us,
                       const float* __restrict__ Wg1, const float* __restrict__ bg1,
                       const float* __restrict__ Wg2, const float* __restrict__ bg2,
                       float* __restrict__ gvec) {
  __shared__ float red[256];
  __shared__ float gl[HH];
  const int t   = threadIdx.x;
  const int b   = blockIdx.x;
  const int h   = t & 63;
  const int grp = t >> 6;        // 0..3
  const float w0 = Wg1[h], w1 = Wg1[HH + h], bb = bg1[h];
  float acc = 0.0f;
  for (int p = grp; p < NN; p += 4) {
    float hv = xs[b * NN + p] * w0 + us[b * NN + p] * w1 + bb;
    acc += hv > 0.0f ? hv : 0.0f;
  }
  red[t] = acc;
  __syncthreads();
  if (t < HH)
    gl[t] = (red[t] + red[64 + t] + red[128 + t] + red[192 + t]) * (1.0f / (float)NN);
  __syncthreads();
  if (t < DD) {
    float a = 0.0f;
    for (int hh = 0; hh < HH; ++hh) a += gl[hh] * Wg2[hh * DD + t];
    gvec[b * DD + t] = a + bg2[t];
  }
}

// ---------------------------------------------------------------------------
// Kernel 2: local encoder with WMMA f32 16x16x4 for the 64x64 hidden GEMM.
//   16 (b,m) pairs per block; each pair = 16 rows (its K=16 neighbors).
//   h1 stored transposed in LDS, W2 staged in LDS. Row-mean of relu(h2)
//   accumulated via ds_add_f32, then (mean @ W3) + b3 + gvec.
// ---------------------------------------------------------------------------
__global__ __launch_bounds__(256)
void sgon_encode_kernel(const float* __restrict__ xs, const float* __restrict__ us,
                        const float* __restrict__ centers,
                        const float* __restrict__ W1, const float* __restrict__ b1,
                        const float* __restrict__ W2, const float* __restrict__ b2,
                        const float* __restrict__ W3, const float* __restrict__ b3,
                        const int*   __restrict__ idx,
                        const float* __restrict__ gvec,
                        float* __restrict__ c0_out) {
  extern __shared__ float smem[];
  float* h1t  = smem;                 // [64][256]  h1 transposed (col j, row r)
  float* w2s  = h1t + HH * 256;       // [64][64]
  float* hbar = w2s + HH * HH;        // [16][64]   sum_k relu(h2)
  const int t   = threadIdx.x;
  const int pi0 = blockIdx.x * 16;    // first (b,m) pair of this block

  for (int i = t; i < 16 * HH; i += 256) hbar[i] = 0.0f;
  for (int i = t; i < HH * HH; i += 256) w2s[i] = W2[i];

  // h1 for row t (pair pi0 + t/16, neighbor k = t%16)
  {
    const int pi = pi0 + (t >> 4);
    const int b  = pi >> 8;           // / MM
    const int m  = pi & 255;
    const int k  = t & 15;
    const int s  = idx[m * KK + k];
    const float x   = xs[b * NN + s];
    const float u   = us[b * NN + s];
    const float rel = (x - centers[m]) / RADIUSF;
    #pragma unroll 8
    for (int j = 0; j < HH; ++j) {
      float hv = rel * W1[j] + u * W1[HH + j] + b1[j];
      h1t[j * 256 + t] = hv > 0.0f ? hv : 0.0f;
    }
  }
  __syncthreads();

  const int wave = t >> 5;
  const int lane = t & 31;
  const int hl   = lane >> 4;         // half-wave select (K split)
  const int l15  = lane & 15;

  // 64 output tiles (16 pairs x 4 col-tiles); 8 per wave.
  for (int tile = 0; tile < 8; ++tile) {
    const int tid8    = wave * 8 + tile;
    const int pLocal  = tid8 >> 2;         // 0..15
    const int ct      = tid8 & 3;          // col tile
    const int rowBase = pLocal * 16;
    const int n0      = ct * 16;
    v8f acc = {};
    #pragma unroll 4
    for (int k0 = 0; k0 < HH; k0 += 4) {
      const int ka = k0 + hl * 2;
      v2f a, bf;
      // A 16x4 f32: lanes 0-15 hold K=k0,k0+1 ; lanes 16-31 hold K=k0+2,k0+3
      a.x  = h1t[(ka + 0) * 256 + rowBase + l15];
      a.y  = h1t[(ka + 1) * 256 + rowBase + l15];
      // B 4x16 f32: same K split, lane&15 = column
      bf.x = w2s[(ka + 0) * HH + n0 + l15];
      bf.y = w2s[(ka + 1) * HH + n0 + l15];
      acc = __builtin_amdgcn_wmma_f32_16x16x4_f32(
          false, a, false, bf, (short)0, acc, false, false);
    }
    // bias + relu + partial sum over the 8 rows this lane holds
    const int col = n0 + l15;
    const float bb = b2[col];
    float part = 0.0f;
    #pragma unroll
    for (int v = 0; v < 8; ++v) {
      float hv = acc[v] + bb;
      part += hv > 0.0f ? hv : 0.0f;
    }
    atomicAdd(&hbar[pLocal * HH + col], part);   // ds_add_f32
  }
  __syncthreads();

  // c0 = (hbar/16) @ W3 + b3 + gvec
  for (int i = t; i < 16 * DD; i += 256) {
    const int p  = i >> 5;
    const int d  = i & 31;
    float a = 0.0f;
    #pragma unroll 8
    for (int h = 0; h < HH; ++h) a += hbar[p * HH + h] * W3[h * DD + d];
    const int pi = pi0 + p;
    const int b  = pi >> 8;
    c0_out[pi * DD + d] = a * (1.0f / (float)KK) + b3[d] + gvec[b * DD + d];
  }
}

// ---------------------------------------------------------------------------
// Kernel 3: per-batch CG glue, fully LDS-resident (1 workgroup per batch).
//   A(v) = v + LAM * L(v), chain graph src=e, dst=e+1.
//   Pass 1: edge residuals r_e = Rs[e] v[e] - Rd[e] v[e+1]   (255x16)
//   Pass 2: L(v)[m] = Rs[m]^T r_m - Rd[m-1]^T r_{m-1}        (gather, no conflicts)
// ---------------------------------------------------------------------------
__device__ inline void sgon_applyA(const float* __restrict__ v, float* __restrict__ AP,
                                   float* __restrict__ RE,
                                   const float* __restrict__ Rsrc,
                                   const float* __restrict__ Rdst, int t) {
  for (int i = t; i < (MM - 1) * RR; i += 512) {
    const int e = i >> 4;
    const float* rs = Rsrc + (size_t)i * DD;
    const float* rd = Rdst + (size_t)i * DD;
    const float* vs = v + e * DD;
    const float* vd = v + (e + 1) * DD;
    float acc = 0.0f;
    #pragma unroll 8
    for (int d = 0; d < DD; ++d) acc += rs[d] * vs[d] - rd[d] * vd[d];
    RE[i] = acc;
  }
  __syncthreads();
  for (int i = t; i < MM * DD; i += 512) {
    const int m = i >> 5;
    const int d = i & 31;
    float acc = 0.0f;
    if (m <= MM - 2) {
      const float* rs = Rsrc + ((size_t)m * RR) * DD + d;
      const float* re = RE + m * RR;
      #pragma unroll
      for (int rr = 0; rr < RR; ++rr) acc += rs[rr * DD] * re[rr];
    }
    if (m >= 1) {
      const float* rd = Rdst + ((size_t)(m - 1) * RR) * DD + d;
      const float* re = RE + (m - 1) * RR;
      #pragma unroll
      for (int rr = 0; rr < RR; ++rr) acc -= rd[rr * DD] * re[rr];
    }
    AP[i] = v[i] + LAMBDA * acc;
  }
  __syncthreads();
}

__device__ inline float sgon_dot(const float* __restrict__ a, const float* __restrict__ b,
                                 float* __restrict__ RED, int t) {
  float local = 0.0f;
  for (int i = t; i < MM * DD; i += 512) local += a[i] * b[i];
  RED[t] = local;
  __syncthreads();
  for (int s = 256; s > 0; s >>= 1) {
    if (t < s) RED[t] += RED[t + s];
    __syncthreads();
  }
  const float r = RED[0];
  __syncthreads();
  return r;
}

__global__ __launch_bounds__(512)
void sgon_cg_kernel(const float* __restrict__ Rsrc, const float* __restrict__ Rdst,
                    const float* __restrict__ c0, float* __restrict__ cout) {
  extern __shared__ float smem[];
  float* X   = smem;               // 8192
  float* Rv  = X + MM * DD;        // 8192
  float* P   = Rv + MM * DD;       // 8192
  float* AP  = P + MM * DD;        // 8192
  float* RE  = AP + MM * DD;       // 4096 (4080 used)
  float* RED = RE + 4096;          // 512
  const int t = threadIdx.x;
  const int b = blockIdx.x;

  for (int i = t; i < MM * DD; i += 512) X[i] = c0[(size_t)b * MM * DD + i];
  __syncthreads();
  sgon_applyA(X, AP, RE, Rsrc, Rdst, t);
  for (int i = t; i < MM * DD; i += 512) {
    const float r = X[i] - AP[i];
    Rv[i] = r;
    P[i]  = r;
  }
  __syncthreads();
  float rs = sgon_dot(Rv, Rv, RED, t);

  for (int it = 0; it < CG_ITERS; ++it) {
    sgon_applyA(P, AP, RE, Rsrc, Rdst, t);
    const float pAp   = sgon_dot(P, AP, RED, t);
    const float alpha = rs / (pAp + 1e-12f);
    for (int i = t; i < MM * DD; i += 512) {
      X[i]  += alpha * P[i];
      Rv[i] -= alpha * AP[i];
    }
    __syncthreads();
    const float rsn  = sgon_dot(Rv, Rv, RED, t);
    const float beta = rsn / (rs + 1e-12f);
    for (int i = t; i < MM * DD; i += 512) P[i] = Rv[i] + beta * P[i];
    __syncthreads();
    rs = rsn;
  }
  for (int i = t; i < MM * DD; i += 512) cout[(size_t)b * MM * DD + i] = X[i];
}

// ---------------------------------------------------------------------------
// Kernel 4: sparse decode. Each q has exactly 3 candidate centers
//   m in {floor(u)-1, floor(u), floor(u)+1}, u=(q+0.5)*M/Q; w==0 elsewhere.
//   s_pred[b,q] = sum_m w[m,q] * dot(phi_q[m,q,:], c[b,m,:])
// ---------------------------------------------------------------------------
__global__ __launch_bounds__(256)
void sgon_decode_kernel(const float* __restrict__ phi, const float* __restrict__ w,
                        const float* __restrict__ c, float* __restrict__ s_out) {
  __shared__ float cs[6 * BB * DD];     // c slice: 6 centers x 8 batches x 32
  const int t   = threadIdx.x;
  const int q0  = blockIdx.x * 256;
  const int mlo = (q0 >> 6) - 1;        // (q0+0.5)/64 floored, minus 1
  for (int i = t; i < 6 * BB * DD; i += 256) {
    const int mi = i >> 8;
    const int b  = (i >> 5) & 7;
    const int d  = i & 31;
    const int mm = mlo + mi;
    cs[i] = (mm >= 0 && mm < MM) ? c[((size_t)b * MM + mm) * DD + d] : 0.0f;
  }
  __syncthreads();

  const int q = q0 + t;
  const float u = ((float)q + 0.5f) * (1.0f / 64.0f);
  const int mc = (int)floorf(u);
  if (mc + 1 >= 0 && mc + 1 < MM)
    __builtin_prefetch(phi + ((size_t)(mc + 1) * QQ + q) * DD, 0, 0);  // global_prefetch_b8

  float acc[BB];
  #pragma unroll
  for (int b = 0; b < BB; ++b) acc[b] = 0.0f;

  for (int mm = mc - 1; mm <= mc + 1; ++mm) {
    if (mm < 0 || mm >= MM) continue;
    const float wv = w[(size_t)mm * QQ + q];
    const float4* pq = (const float4*)(phi + ((size_t)mm * QQ + q) * DD);
    float4 ph[8];
    #pragma unroll
    for (int j = 0; j < 8; ++j) ph[j] = pq[j];
    const int mi = mm - mlo;
    #pragma unroll
    for (int b = 0; b < BB; ++b) {
      const float* cb = &cs[(mi * BB + b) * DD];
      float ds = 0.0f;
      #pragma unroll
      for (int j = 0; j < 8; ++j) {
        ds += ph[j].x * cb[4 * j] + ph[j].y * cb[4 * j + 1] +
              ph[j].z * cb[4 * j + 2] + ph[j].w * cb[4 * j + 3];
      }
      acc[b] += wv * ds;
    }
  }
  #pragma unroll
  for (int b = 0; b < BB; ++b) s_out[(size_t)b * QQ + q] = acc[b];
}

// ---------------------------------------------------------------------------
extern "C" void kernel_launch(void* const* d_in, const int* in_sizes, int n_in,
                              void* d_out, int out_size, void* d_ws, size_t ws_size,
                              hipStream_t stream) {
  (void)in_sizes; (void)n_in; (void)out_size; (void)ws_size;
  const float* xs      = (const float*)d_in[0];
  const float* us      = (const float*)d_in[1];
  const float* phi_q   = (const float*)d_in[2];
  const float* w       = (const float*)d_in[3];
  const float* centers = (const float*)d_in[4];
  const float* R_src   = (const float*)d_in[5];
  const float* R_dst   = (const float*)d_in[6];
  const float* W1      = (const float*)d_in[7];
  const float* b1      = (const float*)d_in[8];
  const float* W2      = (const float*)d_in[9];
  const float* b2      = (const float*)d_in[10];
  const float* W3      = (const float*)d_in[11];
  const float* b3      = (const float*)d_in[12];
  const float* Wg1     = (const float*)d_in[13];
  const float* bg1     = (const float*)d_in[14];
  const float* Wg2     = (const float*)d_in[15];
  const float* bg2     = (const float*)d_in[16];
  const int*   idx     = (const int*)d_in[17];

  float* out    = (float*)d_out;
  float* s_pred = out;                       // B*Q
  float* c0     = out + (size_t)BB * QQ;     // B*M*D
  float* cfin   = c0 + (size_t)BB * MM * DD; // B*M*D
  float* gvec   = (float*)d_ws;              // B*D scratch

  // 1) global path
  sgon_gfeat_kernel<<<BB, 256, 0, stream>>>(xs, us, Wg1, bg1, Wg2, bg2, gvec);

  // 2) encoder (WMMA): 2048 pairs / 16 per block = 128 blocks
  const size_t enc_lds = (size_t)(HH * 256 + HH * HH + 16 * HH) * sizeof(float); // 84 KB
  sgon_encode_kernel<<<128, 256, enc_lds, stream>>>(
      xs, us, centers, W1, b1, W2, b2, W3, b3, idx, gvec, c0);

  // 3) CG glue: one workgroup per batch, ~146 KB LDS of the 320 KB WGP pool
  const size_t cg_lds = (size_t)(4 * MM * DD + 4096 + 512) * sizeof(float);
  sgon_cg_kernel<<<BB, 512, cg_lds, stream>>>(R_src, R_dst, c0, cfin);

  // 4) sparse decode: Q/256 = 64 blocks
  sgon_decode_kernel<<<QQ / 256, 256, 0, stream>>>(phi_q, w, cfin, s_pred);
}